// VertexNormalmap_29884382445937
// MI455X (gfx1250) — compile-verified
//
#include <hip/hip_runtime.h>

// Normal-map lookup over a triangle mesh: per hit, gather triangle verts +
// features, compute barycentrics, blend features, perturb + renormalize the
// hit normal. Pure gather/stream workload: bandwidth bound, L2-resident
// lookup tables (vbo/ibo/features = 24MB << 192MB L2).
//
// Strategy for MI455X: one wave32 lane per hit, b96-merged loads/stores,
// minimal VGPRs for max occupancy so L2 gather latency is hidden.

__global__ __launch_bounds__(256, 8)
void VertexNormalmap_kernel(const float* __restrict__ hit_pos,
                            const float* __restrict__ hit_nrm,
                            const int*   __restrict__ hit_prim,
                            const float* __restrict__ vbo,
                            const int*   __restrict__ ibo,
                            const float* __restrict__ feat,
                            float*       __restrict__ out,
                            int n)
{
    int i = blockIdx.x * blockDim.x + threadIdx.x;
    if (i >= n) return;

    // ---- streamed per-hit data (coalesced b96 across the wave) ----
    const float* hp = hit_pos + 3ull * i;
    const float px = hp[0], py = hp[1], pz = hp[2];

    const int t = hit_prim[i];

    // ---- index gather: one b96 from ibo ----
    const int* tp = ibo + 3ull * t;
    const int i0 = tp[0], i1 = tp[1], i2 = tp[2];

    // ---- vertex gathers: three b96 from vbo (L2-resident, 6MB) ----
    const float* v0p = vbo + 3ull * i0;
    const float* v1p = vbo + 3ull * i1;
    const float* v2p = vbo + 3ull * i2;
    const float v0x = v0p[0], v0y = v0p[1], v0z = v0p[2];
    const float v1x = v1p[0], v1y = v1p[1], v1z = v1p[2];
    const float v2x = v2p[0], v2y = v2p[1], v2z = v2p[2];

    // ---- barycentric coordinates ----
    const float e1x = v1x - v0x, e1y = v1y - v0y, e1z = v1z - v0z;
    const float e2x = v2x - v0x, e2y = v2y - v0y, e2z = v2z - v0z;
    const float ppx = px - v0x,  ppy = py - v0y,  ppz = pz - v0z;

    const float d00 = e1x * e1x + e1y * e1y + e1z * e1z;
    const float d01 = e1x * e2x + e1y * e2y + e1z * e2z;
    const float d11 = e2x * e2x + e2y * e2y + e2z * e2z;
    const float d20 = ppx * e1x + ppy * e1y + ppz * e1z;
    const float d21 = ppx * e2x + ppy * e2y + ppz * e2z;

    const float denom = d00 * d11 - d01 * d01;
    const float rden  = __builtin_amdgcn_rcpf(denom);   // v_rcp_f32
    const float bv = (d11 * d20 - d01 * d21) * rden;
    const float bw = (d00 * d21 - d01 * d20) * rden;
    const float bu = 1.0f - bv - bw;

    // ---- feature gathers: three b96 from features (L2-resident, 6MB) ----
    const float* f0p = feat + 3ull * i0;
    const float* f1p = feat + 3ull * i1;
    const float* f2p = feat + 3ull * i2;
    const float nmx = bu * f0p[0] + bv * f1p[0] + bw * f2p[0];
    const float nmy = bu * f0p[1] + bv * f1p[1] + bw * f2p[1];
    const float nmz = bu * f0p[2] + bv * f1p[2] + bw * f2p[2];

    // ---- perturb + renormalize ----
    const float* hnp = hit_nrm + 3ull * i;
    float Nx = hnp[0] + (nmx * 2.0f - 1.0f) * 0.25f;
    float Ny = hnp[1] + (nmy * 2.0f - 1.0f) * 0.25f;
    float Nz = hnp[2] + (nmz * 2.0f - 1.0f) * 0.25f;

    const float rlen = __builtin_amdgcn_rsqf(Nx * Nx + Ny * Ny + Nz * Nz); // v_rsq_f32
    float* op = out + 3ull * i;
    op[0] = Nx * rlen;
    op[1] = Ny * rlen;
    op[2] = Nz * rlen;
}

extern "C" void kernel_launch(void* const* d_in, const int* in_sizes, int n_in,
                              void* d_out, int out_size, void* d_ws, size_t ws_size,
                              hipStream_t stream)
{
    const float* hit_pos  = (const float*)d_in[0];
    const float* hit_nrm  = (const float*)d_in[1];
    const int*   hit_prim = (const int*)  d_in[2];
    const float* vbo      = (const float*)d_in[3];
    const int*   ibo      = (const int*)  d_in[4];
    const float* feat     = (const float*)d_in[5];
    float*       out      = (float*)d_out;

    const int n = in_sizes[2];            // N_HITS (flat count of hit_primIDs)
    const int block = 256;                // 8 wave32s per block
    const int grid  = (n + block - 1) / block;

    VertexNormalmap_kernel<<<grid, block, 0, stream>>>(
        hit_pos, hit_nrm, hit_prim, vbo, ibo, feat, out, n);
}